// AdvancedQuantumNeuralNetwork_87428354277457
// MI455X (gfx1250) — compile-verified
//
#include <hip/hip_runtime.h>
#include <math.h>

#define NQ      12
#define DIM     4096        // 2^12 amplitudes
#define NLAYERS 6
#define BATCH   512
#define THREADS 128         // 4 waves (wave32)

typedef float v2f __attribute__((ext_vector_type(2)));
typedef float v8f __attribute__((ext_vector_type(8)));

struct __align__(16) Smem {
    float sre[DIM];         // state, real plane   (16 KB)
    float sim[DIM];         // state, imag plane   (16 KB)
    float u16r[256];        // fused 4-qubit gate, real  (U16[r*16+c])
    float u16i[256];        // fused 4-qubit gate, imag
    float gm[NQ][8];        // per-wire 2x2 gate: (rb,cb) -> gm[(rb*2+cb)*2 + {re,im}]
    float red[THREADS];     // reduction buffer
};

__global__ __launch_bounds__(THREADS)
void qnn12_kernel(const float* __restrict__ x,
                  const float* __restrict__ w,
                  const float* __restrict__ bias,
                  float* __restrict__ out,
                  long long x_row_stride)
{
    __shared__ Smem sm;
    const int tid = threadIdx.x;
    const int b   = blockIdx.x;

    // ---------------- init |0...0> ----------------
    for (int i = tid; i < DIM; i += THREADS) { sm.sre[i] = 0.0f; sm.sim[i] = 0.0f; }
    if (tid == 0) sm.sre[0] = 1.0f;

    // ---------------- AngleEmbedding RX(feats[b][j]) ----------------
    if (tid < NQ) {
        float half = 0.5f * x[(long long)b * x_row_stride + tid];
        sm.gm[tid][0] = cosf(half);   // c
        sm.gm[tid][1] = sinf(half);   // s   (gate = [[c,-is],[-is,c]])
    }
    __syncthreads();

    for (int j = 0; j < NQ; ++j) {                  // wire j -> bit 11-j
        const int bit = 11 - j;
        const float c = sm.gm[j][0], s = sm.gm[j][1];
        const int stride = 1 << bit, lo = stride - 1;
        for (int p = tid; p < DIM / 2; p += THREADS) {
            int i0 = ((p >> bit) << (bit + 1)) | (p & lo);
            int i1 = i0 | stride;
            float r0 = sm.sre[i0], m0 = sm.sim[i0];
            float r1 = sm.sre[i1], m1 = sm.sim[i1];
            sm.sre[i0] = c * r0 + s * m1;           // c*s0 - i*s*s1
            sm.sim[i0] = c * m0 - s * r1;
            sm.sre[i1] = c * r1 + s * m0;           // -i*s*s0 + c*s1
            sm.sim[i1] = c * m1 - s * r0;
        }
        __syncthreads();
    }

    // ---------------- layers ----------------
    for (int L = 0; L < NLAYERS; ++L) {
        // Rot(phi,theta,omega) 2x2 matrices for all 12 wires
        if (tid < NQ) {
            const float* wv = w + ((long long)L * NQ + tid) * 3;
            float phi = wv[0], th = wv[1], om = wv[2];
            float hs = 0.5f * (phi + om), hd = 0.5f * (phi - om);
            float ct = cosf(0.5f * th), st = sinf(0.5f * th);
            float chs = cosf(hs), shs = sinf(hs);
            float chd = cosf(hd), shd = sinf(hd);
            float* g = sm.gm[tid];
            g[0] =  ct * chs;  g[1] = -ct * shs;    // a = e^{-i hs} ct
            g[2] = -st * chd;  g[3] = -st * shd;    // b = -e^{+i hd} st
            g[4] =  st * chd;  g[5] = -st * shd;    // c = e^{-i hd} st
            g[6] =  ct * chs;  g[7] =  ct * shs;    // d = e^{+i hs} ct
        }
        __syncthreads();

        // wires 0..7 (bits 11..4): pairwise butterflies in LDS
        for (int j = 0; j < 8; ++j) {
            const int bit = 11 - j;
            const float are = sm.gm[j][0], aim = sm.gm[j][1];
            const float bre = sm.gm[j][2], bim = sm.gm[j][3];
            const float cre = sm.gm[j][4], cim = sm.gm[j][5];
            const float dre = sm.gm[j][6], dim = sm.gm[j][7];
            const int stride = 1 << bit, lo = stride - 1;
            for (int p = tid; p < DIM / 2; p += THREADS) {
                int i0 = ((p >> bit) << (bit + 1)) | (p & lo);
                int i1 = i0 | stride;
                float r0 = sm.sre[i0], m0 = sm.sim[i0];
                float r1 = sm.sre[i1], m1 = sm.sim[i1];
                sm.sre[i0] = are * r0 - aim * m0 + bre * r1 - bim * m1;
                sm.sim[i0] = are * m0 + aim * r0 + bre * m1 + bim * r1;
                sm.sre[i1] = cre * r0 - cim * m0 + dre * r1 - dim * m1;
                sm.sim[i1] = cre * m0 + cim * r0 + dre * m1 + dim * r1;
            }
            __syncthreads();
        }

        // fuse wires 8..11 (bits 3..0): U16 = U8 (x) U9 (x) U10 (x) U11
        for (int e = tid; e < 256; e += THREADS) {
            int r = e >> 4, k = e & 15;
            float pr = 1.0f, pi = 0.0f;
#pragma unroll
            for (int q = 0; q < 4; ++q) {
                int rb = (r >> (3 - q)) & 1, cb = (k >> (3 - q)) & 1;
                const float* g = sm.gm[8 + q];
                float er = g[(rb * 2 + cb) * 2 + 0];
                float ei = g[(rb * 2 + cb) * 2 + 1];
                float nr = pr * er - pi * ei;
                float ni = pr * ei + pi * er;
                pr = nr; pi = ni;
            }
            sm.u16r[e] = pr; sm.u16i[e] = pi;
        }
        __syncthreads();

        // apply fused gate: S(256x16) <- S * U16^T via V_WMMA_F32_16X16X4_F32
        {
            const int lane = tid & 31;
            const int wave = __builtin_amdgcn_readfirstlane(tid >> 5); // SGPR: 0..3
            const int n    = lane & 15;      // col (and A row within tile)
            const int kh   = lane >> 4;      // K half select

            // B fragments (U16^T) are tile-invariant: hoist out of the tile loop
            v2f brf[4], bif[4], nbif[4];
#pragma unroll
            for (int kk = 0; kk < 4; ++kk) {
                const int k0 = kk * 4 + kh * 2;
                brf[kk].x  = sm.u16r[n * 16 + k0];
                brf[kk].y  = sm.u16r[n * 16 + k0 + 1];
                bif[kk].x  = sm.u16i[n * 16 + k0];
                bif[kk].y  = sm.u16i[n * 16 + k0 + 1];
                nbif[kk].x = -bif[kk].x;
                nbif[kk].y = -bif[kk].y;
            }

#pragma unroll
            for (int tt = 0; tt < 4; ++tt) {          // uniform literal trip count
                const int t   = tt * 4 + wave;        // tile of 16 blocks
                const int blk = t * 16 + n;           // this lane's A row (block)
                v8f cr = {}; v8f ci = {};
#pragma unroll
                for (int kk = 0; kk < 4; ++kk) {
                    const int k0 = kk * 4 + kh * 2;
                    v2f ar, ai;
                    ar.x = sm.sre[blk * 16 + k0];
                    ar.y = sm.sre[blk * 16 + k0 + 1];
                    ai.x = sm.sim[blk * 16 + k0];
                    ai.y = sm.sim[blk * 16 + k0 + 1];
                    // Cr += Ar*Br - Ai*Bi ; Ci += Ar*Bi + Ai*Br
                    cr = __builtin_amdgcn_wmma_f32_16x16x4_f32(false, ar, false, brf[kk],  (short)0, cr, false, false);
                    cr = __builtin_amdgcn_wmma_f32_16x16x4_f32(false, ai, false, nbif[kk], (short)0, cr, false, false);
                    ci = __builtin_amdgcn_wmma_f32_16x16x4_f32(false, ar, false, bif[kk],  (short)0, ci, false, false);
                    ci = __builtin_amdgcn_wmma_f32_16x16x4_f32(false, ai, false, brf[kk],  (short)0, ci, false, false);
                }
#pragma unroll
                for (int v = 0; v < 8; ++v) {         // D: VGPR v -> row v (+8 hi lanes)
                    int m = t * 16 + v + kh * 8;
                    sm.sre[m * 16 + n] = cr[v];
                    sm.sim[m * 16 + n] = ci[v];
                }
            }
        }
        __syncthreads();

        // CNOT ring: ctrl bit 11-j, tgt bit 11-((j+1)%12); swap t-pair where c=1
        for (int j = 0; j < NQ; ++j) {
            const int cb = 11 - j;
            const int tb = 11 - ((j + 1) % NQ);
            const int b1 = cb < tb ? cb : tb;
            const int b2 = cb < tb ? tb : cb;
            for (int q = tid; q < DIM / 4; q += THREADS) {
                int low  = q & ((1 << b1) - 1);
                int mid  = (q >> b1) & ((1 << (b2 - b1 - 1)) - 1);
                int high = q >> (b2 - 1);
                int i0 = (high << (b2 + 1)) | (mid << (b1 + 1)) | low | (1 << cb);
                int i1 = i0 | (1 << tb);
                float tr = sm.sre[i0]; sm.sre[i0] = sm.sre[i1]; sm.sre[i1] = tr;
                float ti = sm.sim[i0]; sm.sim[i0] = sm.sim[i1]; sm.sim[i1] = ti;
            }
            __syncthreads();
        }
    }

    // ---------------- <Z_0> = P(bit11=0) - P(bit11=1) ----------------
    float local = 0.0f;
    for (int i = tid; i < DIM; i += THREADS) {
        float p = sm.sre[i] * sm.sre[i] + sm.sim[i] * sm.sim[i];
        local += (i & 2048) ? -p : p;
    }
    sm.red[tid] = local;
    __syncthreads();
    for (int s = THREADS / 2; s > 0; s >>= 1) {
        if (tid < s) sm.red[tid] += sm.red[tid + s];
        __syncthreads();
    }
    if (tid == 0) out[b] = sm.red[0] + bias[0];
}

extern "C" void kernel_launch(void* const* d_in, const int* in_sizes, int n_in,
                              void* d_out, int out_size, void* d_ws, size_t ws_size,
                              hipStream_t stream)
{
    (void)n_in; (void)out_size; (void)d_ws; (void)ws_size;
    const float* x    = (const float*)d_in[0];   // (512, 3, 224, 224)
    const float* w    = (const float*)d_in[1];   // (6, 12, 3)
    const float* bias = (const float*)d_in[2];   // scalar
    float* out = (float*)d_out;                  // (512, 1)
    long long row = (long long)in_sizes[0] / BATCH;   // 150528
    qnn12_kernel<<<BATCH, THREADS, 0, stream>>>(x, w, bias, out, row);
}